// NearestNeighbor_55181739819637
// MI455X (gfx1250) — compile-verified
//
#include <hip/hip_runtime.h>

// ---------------------------------------------------------------------------
// Fused batched NN-matcher for MI455X (gfx1250, wave32).
//   pass 1: top-2 over m of sim[b,n,:]  (d0 rows vs d1 cols)
//   pass 2: top-2 over n of sim[b,:,m]  == same kernel with d0/d1 swapped
//   pass 3: ratio test + mutual check
// sim tiles are produced by V_WMMA_F32_16X16X4_F32 (full fp32, matching the
// fp32 reference) and reduced in-register: never written to HBM.
// ---------------------------------------------------------------------------

typedef float v2f __attribute__((ext_vector_type(2)));
typedef float v8f __attribute__((ext_vector_type(8)));

constexpr int   B    = 8;
constexpr int   D    = 256;
constexpr int   N    = 4096;
constexpr int   M    = 4096;
constexpr int   ROWS = 32;      // rows per workgroup = 2 waves x 16
constexpr int   STRA = 260;     // padded LDS row stride (dwords): even (b64-
                                // aligned frag loads), 260%64=4 spreads banks
constexpr float RT2  = 0.64f;   // Lowe ratio^2 = 0.8^2

// For each (b, row): top-2 similarity over all M columns.
// X: [B][D][N] row-side descriptors, Y: [B][D][M] column-side descriptors.
__global__ __launch_bounds__(64)
void top2_rows_kernel(const float* __restrict__ X, const float* __restrict__ Y,
                      float* __restrict__ o_v1, float* __restrict__ o_v2,
                      int* __restrict__ o_i1)
{
    __shared__ float lds_a[ROWS * STRA];   // A strip, [row][k], 33.3 KB
    __shared__ float lds_b[D * 16];        // B tile,  [k][col], 16 KB

    const int t    = threadIdx.x;
    const int wid  = t >> 5;               // wave id (0..1)
    const int lane = t & 31;
    const int lr   = lane & 15;            // col within tile / row within frag
    const int kh   = lane >> 4;            // half-wave id
    const int bb   = blockIdx.y;
    const int n0   = blockIdx.x * ROWS;

    // ---- stage A strip: X[bb, k, n0+row] -> lds_a[row*STRA + k] -----------
    const float* Xb = X + (size_t)bb * D * N;
    for (int j = 0; j < (ROWS * D) / 64; ++j) {        // 128 iters
        int idx = j * 64 + t;
        int k   = idx >> 5;                            // ROWS == 32
        int row = idx & (ROWS - 1);                    // coalesced global read
        lds_a[row * STRA + k] = Xb[(size_t)k * N + n0 + row];
    }
    __syncthreads();

    // Running per-row top-2, kept lane-local (this lane's slot = cols == lr
    // mod 16).  WMMA C/D layout: acc[v] of lane L holds sim(row v+8*kh, col
    // m0+lr), so rows are compile-time indexed and no shuffles are needed in
    // the main loop.  Top-2(union of slots) == merge of per-slot top-2s.
    float rv1[8], rv2[8];
    int   ri1[8];
#pragma unroll
    for (int v = 0; v < 8; ++v) { rv1[v] = -3.0e38f; rv2[v] = -3.0e38f; ri1[v] = 0; }

    const float* Yb   = Y + (size_t)bb * D * M;
    const float* arow = &lds_a[(16 * wid + lr) * STRA + 2 * kh];

    for (int m0 = 0; m0 < M; m0 += 16) {
        // ---- stage B tile: Y[bb, k, m0..m0+15] -> lds_b[k*16 + c] ---------
#pragma unroll
        for (int j = 0; j < (D * 16 / 4) / 64; ++j) {  // 16 x float4 / thread
            int idx4 = j * 64 + t;
            int k    = idx4 >> 2;
            int c4   = (idx4 & 3) * 4;
            const float4 g = *(const float4*)(Yb + (size_t)k * M + m0 + c4);
            *(float4*)&lds_b[k * 16 + c4] = g;
        }
        __syncthreads();

        // ---- 16x16 fp32 sim tile: 64 x v_wmma_f32_16x16x4_f32 -------------
        v8f acc = {};
#pragma unroll 8
        for (int kk = 0; kk < D / 4; ++kk) {
            // A frag (16x4): lanes 0-15 hold K={0,1}, lanes 16-31 K={2,3}
            v2f a = *(const v2f*)(arow + 4 * kk);
            // B frag (4x16): vgpr0 = rows K0/K1 across half-waves, vgpr1 = K2/K3
            v2f b;
            b.x = lds_b[(4 * kk + kh) * 16 + lr];
            b.y = lds_b[(4 * kk + 2 + kh) * 16 + lr];
            acc = __builtin_amdgcn_wmma_f32_16x16x4_f32(
                      false, a, false, b, (short)0, acc, false, false);
        }

        // ---- lane-local top-2 merge (5 VALU per row, no shuffles) ---------
        const int midx = m0 + lr;
#pragma unroll
        for (int v = 0; v < 8; ++v) {
            float x     = acc[v];
            float loser = fminf(x, rv1[v]);            // loser of the top spot
            bool  gt    = x > rv1[v];                  // strict > keeps lowest
            rv2[v] = fmaxf(rv2[v], loser);             //   index on ties (jax)
            rv1[v] = fmaxf(rv1[v], x);
            ri1[v] = gt ? midx : ri1[v];
        }
        __syncthreads();                               // before next B stage
    }

    // ---- final cross-lane merge inside each 16-lane half ------------------
#pragma unroll
    for (int v = 0; v < 8; ++v) {
        float a1 = rv1[v], a2 = rv2[v];
        int   ai = ri1[v];
#pragma unroll
        for (int s = 1; s < 16; s <<= 1) {
            float b1 = __shfl_xor(a1, s, 32);
            float b2 = __shfl_xor(a2, s, 32);
            int   bi = __shfl_xor(ai, s, 32);
            bool take = (b1 > a1) || ((b1 == a1) && (bi < ai)); // tie -> low idx
            a2 = fmaxf(fmaxf(a2, b2), fminf(a1, b1));  // 2nd of union
            a1 = fmaxf(a1, b1);
            ai = take ? bi : ai;
        }
        rv1[v] = a1; rv2[v] = a2; ri1[v] = ai;
    }
    if (lr == 0) {                                     // lanes 0 & 16 write
#pragma unroll
        for (int v = 0; v < 8; ++v) {
            int    row = n0 + 16 * wid + v + 8 * kh;
            size_t o   = (size_t)bb * N + row;
            o_v1[o] = rv1[v];
            o_v2[o] = rv2[v];
            o_i1[o] = ri1[v];
        }
    }
}

// Ratio test + mutual check + scores.
__global__ __launch_bounds__(256)
void nn_finalize_kernel(const float* __restrict__ v1_0, const float* __restrict__ v2_0,
                        const int* __restrict__ i1_0,
                        const float* __restrict__ v1_1, const float* __restrict__ v2_1,
                        const int* __restrict__ i1_1,
                        int* __restrict__ out_matches, float* __restrict__ out_scores)
{
    int i = blockIdx.x * blockDim.x + threadIdx.x;
    if (i >= B * N) return;
    int bb = i / N;
    int ii = i - bb * N;

    float v1 = v1_0[i], v2 = v2_0[i];
    int   j1 = i1_0[i];
    // dist = 2(1-sim); accept iff dist1 <= 0.64 * dist2
    bool ok0 = (1.0f - v1) <= RT2 * (1.0f - v2);
    int  m0  = ok0 ? j1 : -1;

    int mm = -1;                                       // matches1[m0]
    if (m0 >= 0) {
        size_t j   = (size_t)bb * M + m0;
        bool   ok1 = (1.0f - v1_1[j]) <= RT2 * (1.0f - v2_1[j]);
        mm = ok1 ? i1_1[j] : -1;
    }
    out_matches[i] = (m0 > -1 && mm == ii) ? m0 : -1;
    out_scores[i]  = ok0 ? (v1 + 1.0f) * 0.5f : 0.0f;  // scores pre-mutual
}

extern "C" void kernel_launch(void* const* d_in, const int* in_sizes, int n_in,
                              void* d_out, int out_size, void* d_ws, size_t ws_size,
                              hipStream_t stream)
{
    const float* d0 = (const float*)d_in[0];   // [B][D][N]
    const float* d1 = (const float*)d_in[1];   // [B][D][M]

    // workspace: 6 arrays of B*N 4-byte values (768 KB total)
    float* ws   = (float*)d_ws;
    float* v1_0 = ws + 0 * (size_t)B * N;
    float* v2_0 = ws + 1 * (size_t)B * N;
    int*   i1_0 = (int*)(ws + 2 * (size_t)B * N);
    float* v1_1 = ws + 3 * (size_t)B * N;
    float* v2_1 = ws + 4 * (size_t)B * N;
    int*   i1_1 = (int*)(ws + 5 * (size_t)B * N);

    dim3 grid(N / ROWS, B);
    // pass 1: rows = d0, cols = d1  -> matches0 candidates
    top2_rows_kernel<<<grid, 64, 0, stream>>>(d0, d1, v1_0, v2_0, i1_0);
    // pass 2: rows = d1, cols = d0  -> matches1 candidates (transposed sim)
    top2_rows_kernel<<<grid, 64, 0, stream>>>(d1, d0, v1_1, v2_1, i1_1);

    // output: [ matches0 : B*N int32 | scores0 : B*N float32 ]
    int tot = B * N;
    nn_finalize_kernel<<<(tot + 255) / 256, 256, 0, stream>>>(
        v1_0, v2_0, i1_0, v1_1, v2_1, i1_1,
        (int*)d_out, (float*)d_out + (size_t)B * N);
}